// DecomposedConvolutionalAttention_24532853194909
// MI455X (gfx1250) — compile-verified
//
#include <hip/hip_runtime.h>
#include <math.h>

typedef __attribute__((ext_vector_type(2))) float v2f;
typedef __attribute__((ext_vector_type(8))) float v8f;

#define B_    8
#define CIN   128
#define PDIM  64
#define H_    256
#define W_    256
#define LK    13
#define KK2   (LK*LK)
#define HALO  6           // LK/2
#define TILE  16
#define TIN   28          // TILE + 2*HALO
#define NPOS  784         // TIN*TIN == 49 * 16 exactly
#define XIN_STRIDE  68    // pos-major channel stride: conflict-free b64 A-frag reads
#define XMIX_STRIDE 788   // lane L -> bank offset L*20 mod 64, all distinct

#define HAS_ASYNC_LDS __has_builtin(__builtin_amdgcn_global_load_async_to_lds_b32)

// ---------------- Kernel 1: global average pool over H*W for x1 ----------------
__global__ void pool_kernel(const float* __restrict__ x, float* __restrict__ pooled) {
    int bc = blockIdx.x;               // b*64 + c, 512 blocks
    int b = bc >> 6, c = bc & 63;
    const float4* p = (const float4*)(x + (((size_t)(b * CIN + c)) << 16));
    float s = 0.f;
    for (int i = threadIdx.x; i < (H_ * W_ / 4); i += blockDim.x) {
        float4 v = p[i];
        s += v.x + v.y + v.z + v.w;
    }
    __shared__ float red[256];
    red[threadIdx.x] = s;
    __syncthreads();
    for (int off = 128; off > 0; off >>= 1) {
        if (threadIdx.x < off) red[threadIdx.x] += red[threadIdx.x + off];
        __syncthreads();
    }
    if (threadIdx.x == 0) pooled[bc] = red[0] * (1.f / (float)(H_ * W_));
}

// ---------------- Kernel 2: MLP -> dynamic 13x13 depthwise kernels ----------------
__global__ void mlp_kernel(const float* __restrict__ pooled,
                           const float* __restrict__ w1, const float* __restrict__ b1,
                           const float* __restrict__ w2, const float* __restrict__ b2,
                           const float* __restrict__ lk_spatial,
                           float* __restrict__ kernfull) {
    __shared__ float hid[B_ * 16];
    int t = threadIdx.x;
    if (t < B_ * 16) {
        int b = t >> 4, j = t & 15;
        float s = b1[j];
        for (int c = 0; c < PDIM; ++c) s += pooled[b * PDIM + c] * w1[j * PDIM + c];
        // exact GELU: 0.5*x*(1+erf(x/sqrt(2)))
        hid[t] = 0.5f * s * (1.f + erff(s * 0.70710678118654752f));
    }
    __syncthreads();
    int total = B_ * PDIM * KK2;
    for (int idx = t; idx < total; idx += blockDim.x) {
        int b  = idx / (PDIM * KK2);
        int r  = idx - b * (PDIM * KK2);
        int c  = r / KK2;
        int kk = r - c * KK2;
        int ky = kk / LK, kx = kk - ky * LK;
        float v = lk_spatial[c * KK2 + kk];
        if (ky >= 5 && ky < 8 && kx >= 5 && kx < 8) {     // central 3x3 (PAD=5)
            int i = c * 9 + (ky - 5) * 3 + (kx - 5);
            float s = b2[i];
            for (int j = 0; j < 16; ++j) s += hid[b * 16 + j] * w2[i * 16 + j];
            v += s;
        }
        kernfull[idx] = v;
    }
}

// ------- Kernel 3: fused channel-mix (WMMA f32 16x16x4) + depthwise 13x13 -------
__global__ void fused_kernel(const float* __restrict__ x,
                             const float* __restrict__ lkc,      // (64,64): [o*64+c]
                             const float* __restrict__ kernfull, // (8,64,13,13)
                             float* __restrict__ out) {
    extern __shared__ float smem[];
    float* xin  = smem;                     // [784][68]  pos-major x1 halo tile
    float* xmix = smem + NPOS * XIN_STRIDE; // [16][788]  channel-mixed group tile

    int b   = blockIdx.z;
    int ty0 = blockIdx.y * TILE, tx0 = blockIdx.x * TILE;
    int t   = threadIdx.x;

    // ---- Stage x1 28x28 halo tile, all 64 channels; OOB -> 0 (== conv zero pad)
    // Global reads are c-major (consecutive gx => coalesced); LDS store transposed.
#if HAS_ASYNC_LDS
    {
        typedef __attribute__((address_space(1))) int gint;   // global i32
        typedef __attribute__((address_space(3))) int lint;   // LDS i32
        for (int idx = t; idx < PDIM * NPOS; idx += 256) {
            int c  = idx / NPOS;
            int p  = idx - c * NPOS;
            int py = p / TIN, px = p - py * TIN;
            int gy = ty0 - HALO + py, gx = tx0 - HALO + px;
            if ((unsigned)gy < (unsigned)H_ && (unsigned)gx < (unsigned)W_) {
                // async DMA: global -> LDS, no VGPR round-trip (ASYNCcnt)
                __builtin_amdgcn_global_load_async_to_lds_b32(
                    (gint*)(x + (((size_t)(b * CIN + c)) << 16) + gy * W_ + gx),
                    (lint*)(xin + p * XIN_STRIDE + c), 0, 0);
            } else {
                xin[p * XIN_STRIDE + c] = 0.f;   // halo zero-fill
            }
        }
    }
#if __has_builtin(__builtin_amdgcn_s_wait_asynccnt)
    __builtin_amdgcn_s_wait_asynccnt(0);
#else
    asm volatile("s_wait_asynccnt 0" ::: "memory");
#endif
#else
    for (int idx = t; idx < PDIM * NPOS; idx += 256) {
        int c  = idx / NPOS;
        int p  = idx - c * NPOS;
        int py = p / TIN, px = p - py * TIN;
        int gy = ty0 - HALO + py, gx = tx0 - HALO + px;
        float v = 0.f;
        if ((unsigned)gy < (unsigned)H_ && (unsigned)gx < (unsigned)W_)
            v = x[(((size_t)(b * CIN + c)) << 16) + gy * W_ + gx];
        xin[p * XIN_STRIDE + c] = v;
    }
#endif

    int wave = t >> 5, lane = t & 31;
    int mrow = lane & 15;              // M (A/D) or N (B/D-col) index
    int half = lane >> 4;              // 0: K=0,1 / rows 0-7;  1: K=2,3 / rows 8-15
    int mo   = half * 8;
    int ty   = t >> 4, tx = t & 15;    // depthwise: one output pixel per thread

    for (int og = 0; og < PDIM / 16; ++og) {
        __syncthreads();               // xin ready / xmix free for reuse

        // B fragments: B[k=c, n=o] = lkc[(og*16+n)*64 + c], 16x16x4 layout
        v2f bfrag[16];
        #pragma unroll
        for (int k = 0; k < 16; ++k) {
            int c0 = 4 * k + half * 2;
            bfrag[k].x = lkc[(og * 16 + mrow) * PDIM + c0];
            bfrag[k].y = lkc[(og * 16 + mrow) * PDIM + c0 + 1];
        }

        // GEMM: 49 M-tiles of 16 positions, K=64 in 16 steps of 4 (full fp32).
        // A frag = one aligned ds_load_b64 (pos-major layout, stride 68 floats).
        for (int m = wave; m < 49; m += 8) {
            v8f acc = {};
            const v2f* arow = (const v2f*)(xin + (m * 16 + mrow) * XIN_STRIDE);
            #pragma unroll
            for (int k = 0; k < 16; ++k) {
                v2f a = arow[2 * k + half];     // elements c0, c0+1
                acc = __builtin_amdgcn_wmma_f32_16x16x4_f32(
                          false, a, false, bfrag[k], (short)0, acc, false, false);
            }
            // D[M,N] -> xmix[N][m*16+M] (lanes<16: M=r, lanes>=16: M=r+8)
            #pragma unroll
            for (int r = 0; r < 8; ++r)
                xmix[mrow * XMIX_STRIDE + m * 16 + mo + r] = acc[r];
        }
        __syncthreads();               // xmix ready

        // Depthwise 13x13 with dynamic per-(b,channel) kernels
        const float* kf = kernfull + (size_t)(b * PDIM + og * 16) * KK2;
        for (int oc = 0; oc < 16; ++oc) {
            const float* xm = xmix + oc * XMIX_STRIDE + ty * TIN + tx;
            const float* kw = kf + oc * KK2;
            float s = 0.f;
            for (int dy = 0; dy < LK; ++dy) {
                #pragma unroll
                for (int dx = 0; dx < LK; ++dx)
                    s += xm[dy * TIN + dx] * kw[dy * LK + dx];
            }
            out[(((size_t)(b * CIN + og * 16 + oc)) << 16) + (ty0 + ty) * W_ + (tx0 + tx)] = s;
        }
    }
}

// ---------------- Kernel 4: passthrough of x2 (channels 64..127) ----------------
__global__ void copy_x2_kernel(const float* __restrict__ x, float* __restrict__ out) {
    size_t j4 = (size_t)blockIdx.x * blockDim.x + threadIdx.x;     // float4 index
    const size_t per_b4 = (size_t)PDIM * H_ * W_ / 4;              // 1,048,576
    const size_t total4 = (size_t)B_ * per_b4;
    if (j4 >= total4) return;
    size_t b = j4 / per_b4;
    size_t rem = j4 - b * per_b4;
    size_t off4 = (b * CIN + PDIM) * ((size_t)H_ * W_ / 4) + rem;
    ((float4*)out)[off4] = ((const float4*)x)[off4];
}

extern "C" void kernel_launch(void* const* d_in, const int* in_sizes, int n_in,
                              void* d_out, int out_size, void* d_ws, size_t ws_size,
                              hipStream_t stream) {
    const float* x          = (const float*)d_in[0];
    const float* lk_channel = (const float*)d_in[1];
    const float* lk_spatial = (const float*)d_in[2];
    const float* w1         = (const float*)d_in[3];
    const float* b1         = (const float*)d_in[4];
    const float* w2         = (const float*)d_in[5];
    const float* b2         = (const float*)d_in[6];
    float* out = (float*)d_out;

    float* pooled   = (float*)d_ws;             // 512 floats
    float* kernfull = pooled + B_ * PDIM;       // 8*64*169 floats (~346 KB total ws)

    pool_kernel<<<B_ * PDIM, 256, 0, stream>>>(x, pooled);
    mlp_kernel<<<1, 256, 0, stream>>>(pooled, w1, b1, w2, b2, lk_spatial, kernfull);

    size_t smem = (size_t)(NPOS * XIN_STRIDE + 16 * XMIX_STRIDE) * sizeof(float); // 263,680 B
    (void)hipFuncSetAttribute((const void*)fused_kernel,
                              hipFuncAttributeMaxDynamicSharedMemorySize, (int)smem);
    fused_kernel<<<dim3(W_ / TILE, H_ / TILE, B_), 256, smem, stream>>>(
        x, lk_channel, kernfull, out);

    size_t total4 = (size_t)B_ * PDIM * H_ * W_ / 4;
    copy_x2_kernel<<<(int)((total4 + 255) / 256), 256, 0, stream>>>(x, out);
}